// DisagreementFeatureRegularizer_62964220559919
// MI455X (gfx1250) — compile-verified
//
#include <hip/hip_runtime.h>
#include <stdint.h>

// Problem shape (fixed by the reference's setup_inputs)
#define SPATIAL_SHIFT 20
#define SPATIAL (1 << SPATIAL_SHIFT)        // 64*128*128 = 1048576
#define NVOX (2 * SPATIAL)                  // B=2 -> 2097152 voxels
#define NCLS 4
#define NFEAT 32
#define FEAT_ELEMS ((size_t)NVOX * NFEAT)   // 67108864
#define EPSF 1e-8f
#define LOG2F_C 0.6931471805599453f

// kernel-1 pipeline geometry
#define TILE_VOX 1024                       // voxels per tile (4 per thread, 256 threads)
#define NTILES (NVOX / TILE_VOX)            // 2048
#define K1_BLOCKS 512
#define TILES_PER_BLOCK (NTILES / K1_BLOCKS) // 4

// native clang vector type (accepted by __builtin_nontemporal_*)
typedef float vf4 __attribute__((ext_vector_type(4)));
typedef int   vi4 __attribute__((ext_vector_type(4)));

// ---------- order-preserving float <-> uint encoding (for atomic min/max) ----------
__device__ __forceinline__ unsigned fenc(float x) {
  unsigned u = __float_as_uint(x);
  return (u & 0x80000000u) ? ~u : (u | 0x80000000u);
}
__device__ __forceinline__ float fdec(unsigned e) {
  unsigned b = (e & 0x80000000u) ? (e & 0x7FFFFFFFu) : ~e;
  return __uint_as_float(b);
}

// ---------- per-voxel JS divergence over 4 classes ----------
__device__ __forceinline__ float js_dis4(const float a[4], const float b[4]) {
  float ma = fmaxf(fmaxf(a[0], a[1]), fmaxf(a[2], a[3]));
  float mb = fmaxf(fmaxf(b[0], b[1]), fmaxf(b[2], b[3]));
  float ea[4], eb[4];
  float sa = 0.f, sb = 0.f;
#pragma unroll
  for (int c = 0; c < 4; ++c) {
    ea[c] = __expf(a[c] - ma); sa += ea[c];
    eb[c] = __expf(b[c] - mb); sb += eb[c];
  }
  float lsa = __logf(sa), lsb = __logf(sb);
  float isa = 1.f / sa, isb = 1.f / sb;
  float kl = 0.f;
#pragma unroll
  for (int c = 0; c < 4; ++c) {
    float lpa = a[c] - ma - lsa;          // log_softmax A
    float lpb = b[c] - mb - lsb;          // log_softmax B
    float pa = ea[c] * isa;
    float pb = eb[c] * isb;
    float lm = __logf(0.5f * (pa + pb) + EPSF);
    kl += pa * (lpa - lm) + pb * (lpb - lm);
  }
  return 0.5f * kl;
}

// ---------- kernel 0: init global class stats ----------
__global__ void k_init(unsigned* __restrict__ gmin, unsigned* __restrict__ gmax) {
  int t = threadIdx.x;
  if (t < NCLS) { gmin[t] = 0xFFFFFFFFu; gmax[t] = 0u; }
}

// ---------- kernel 1: dis + per-class min/max ----------
// Double-buffered async global->LDS pipeline (CDNA5 ASYNCcnt path):
// while computing tile t out of one 32KB LDS buffer, the async DMA engine fills
// the other buffer with tile t+1 (8 planes x b128 per lane). Async loads
// complete in order, so `s_wait_asynccnt 8` releases exactly the current tile.
template <bool STORE_DIS>
__global__ __launch_bounds__(256) void k_dis_stats(
    const float* __restrict__ lA, const float* __restrict__ lB,
    const int* __restrict__ lab,
    unsigned* __restrict__ gmin, unsigned* __restrict__ gmax,
    float* __restrict__ disbuf) {
  // 2 buffers x 8 planes (4 classes x {A,B}) x 1024 floats = 64 KB LDS
  __shared__ float sbuf[2][8 * TILE_VOX];
  __shared__ unsigned smin[NCLS], smax[NCLS];

  const int tid = threadIdx.x;
  if (tid < NCLS) { smin[tid] = 0xFFFFFFFFu; smax[tid] = 0u; }

  // ---- issue async DMA for one tile into buffer `buf` ----
  auto issue_tile = [&](int buf, int tile) {
    const int v0 = tile * TILE_VOX;
    const int b  = v0 >> SPATIAL_SHIFT;          // tiles never straddle batch
    const int s0 = v0 & (SPATIAL - 1);
    const int e  = s0 + tid * 4;                 // element offset within plane
#pragma unroll
    for (int c = 0; c < NCLS; ++c) {
      unsigned la = (unsigned)(uintptr_t)(void*)&sbuf[buf][c * TILE_VOX + tid * 4];
      unsigned lb = (unsigned)(uintptr_t)(void*)&sbuf[buf][(NCLS + c) * TILE_VOX + tid * 4];
      unsigned long long ga =
          (unsigned long long)(uintptr_t)(lA + (((size_t)(b * NCLS + c)) << SPATIAL_SHIFT) + e);
      unsigned long long gb =
          (unsigned long long)(uintptr_t)(lB + (((size_t)(b * NCLS + c)) << SPATIAL_SHIFT) + e);
      asm volatile("global_load_async_to_lds_b128 %0, %1, off" :: "v"(la), "v"(ga) : "memory");
      asm volatile("global_load_async_to_lds_b128 %0, %1, off" :: "v"(lb), "v"(gb) : "memory");
    }
  };

  issue_tile(0, blockIdx.x);                     // prime the pipeline
  __syncthreads();                               // smin/smax init (overlaps with DMA)

  for (int t = 0; t < TILES_PER_BLOCK; ++t) {
    const int tile = blockIdx.x + t * K1_BLOCKS;
    const int cur  = t & 1;
    const bool more = (t + 1) < TILES_PER_BLOCK;
    if (more) issue_tile(cur ^ 1, blockIdx.x + (t + 1) * K1_BLOCKS);

    // release current tile: with next tile in flight, wait counter <= 8
    if (more) asm volatile("s_wait_asynccnt 8" ::: "memory");
    else      asm volatile("s_wait_asynccnt 0" ::: "memory");

    const long long g4 = (long long)tile * (TILE_VOX / 4) + tid;  // vec4 index
    vi4 l4 = ((const vi4*)lab)[g4];
    int cl[4] = { l4.x & 3, l4.y & 3, l4.z & 3, l4.w & 3 };

    float dis[4];
#pragma unroll
    for (int j = 0; j < 4; ++j) {
      float a[4], bb[4];
#pragma unroll
      for (int c = 0; c < NCLS; ++c) {
        a[c]  = sbuf[cur][c * TILE_VOX + tid * 4 + j];
        bb[c] = sbuf[cur][(NCLS + c) * TILE_VOX + tid * 4 + j];
      }
      dis[j] = js_dis4(a, bb);
    }

    if (STORE_DIS) {
      // dis is re-read by kernel 3: keep regular-temporal so it stays in the
      // 192MB L2 between kernels.
      vf4 dv = { dis[0], dis[1], dis[2], dis[3] };
      ((vf4*)disbuf)[g4] = dv;
    }
#pragma unroll
    for (int j = 0; j < 4; ++j) {
      unsigned e = fenc(dis[j]);
      atomicMin(&smin[cl[j]], e);
      atomicMax(&smax[cl[j]], e);
    }
  }

  __syncthreads();
  if (tid < NCLS) {
    if (smin[tid] != 0xFFFFFFFFu) atomicMin(&gmin[tid], smin[tid]);
    if (smax[tid] != 0u)          atomicMax(&gmax[tid], smax[tid]);
  }
}

// ---------- kernel 2: fold class stats into per-class affine (gamma = slope*dis + bias) ----------
__global__ void k_final(const unsigned* __restrict__ gmin, const unsigned* __restrict__ gmax,
                        const float* __restrict__ ranks,
                        float* __restrict__ slope, float* __restrict__ bias) {
  int c = threadIdx.x;
  if (c >= NCLS) return;
  unsigned mu = gmin[c], xu = gmax[c];
  bool present = (mu != 0xFFFFFFFFu);
  float dmin = present ? fdec(mu) : 0.f;
  float dmax = present ? fdec(xu) : LOG2F_C;
  float inter = 1.0f * (1.f - ranks[c] * (1.f / (NCLS - 1)));  // STRENGTH_INTER = 1
  float sl, bi;
  if (dmax > dmin) {
    float inv = 1.f / (dmax - dmin + EPSF);
    sl = inter * inv;
    bi = 1.f - inter * dmin * inv;
  } else {
    sl = 0.f;
    bi = 1.f + 0.5f * inter;
  }
  slope[c] = sl;
  bias[c]  = bi;
}

// ---------- kernel 3: gamma + broadcast multiply over 32 channels ----------
// The 520MB feature stream has zero reuse: use non-temporal hints (th:NT) so it
// does not sweep the 192MB L2 and evict the dis/labels lines written by kernel 1
// (which this kernel re-reads with regular-temporal loads -> L2 hits).
template <bool HAS_DIS>
__global__ __launch_bounds__(256) void k_apply(
    const float* __restrict__ feat,
    const float* __restrict__ lA, const float* __restrict__ lB,
    const int* __restrict__ lab, const float* __restrict__ disbuf,
    const float* __restrict__ slope, const float* __restrict__ bias,
    float* __restrict__ out) {
  const long long g  = (long long)blockIdx.x * 256 + threadIdx.x;  // one vec4 of voxels
  const long long v0 = g * 4;
  const int b  = (int)(v0 >> SPATIAL_SHIFT);
  const int s0 = (int)(v0 & (SPATIAL - 1));

  float sl[NCLS], bi[NCLS];
#pragma unroll
  for (int c = 0; c < NCLS; ++c) { sl[c] = slope[c]; bi[c] = bias[c]; }

  vi4 l4 = ((const vi4*)lab)[g];               // RT load: expect L2 hit
  int cl[4] = { l4.x & 3, l4.y & 3, l4.z & 3, l4.w & 3 };

  float d[4];
  if (HAS_DIS) {
    vf4 dd = ((const vf4*)disbuf)[g];          // RT load: expect L2 hit
    d[0] = dd.x; d[1] = dd.y; d[2] = dd.z; d[3] = dd.w;
  } else {
    vf4 A4[NCLS], B4[NCLS];
#pragma unroll
    for (int c = 0; c < NCLS; ++c) {
      size_t idx = (((size_t)(b * NCLS + c)) << SPATIAL_SHIFT) + s0;
      A4[c] = ((const vf4*)lA)[idx >> 2];
      B4[c] = ((const vf4*)lB)[idx >> 2];
    }
#pragma unroll
    for (int j = 0; j < 4; ++j) {
      float a[4]  = { A4[0][j], A4[1][j], A4[2][j], A4[3][j] };
      float bb[4] = { B4[0][j], B4[1][j], B4[2][j], B4[3][j] };
      d[j] = js_dis4(a, bb);
    }
  }

  vf4 gam;
#pragma unroll
  for (int j = 0; j < 4; ++j) gam[j] = sl[cl[j]] * d[j] + bi[cl[j]];

  __builtin_nontemporal_store(gam, (vf4*)(out + FEAT_ELEMS) + g);  // gamma output (no reuse)

  const size_t base = (((size_t)b * NFEAT) << SPATIAL_SHIFT) + s0;
#pragma unroll 8
  for (int f = 0; f < NFEAT; ++f) {
    size_t idx = base + ((size_t)f << SPATIAL_SHIFT);
    vf4 x = __builtin_nontemporal_load((const vf4*)feat + (idx >> 2));
    x = x * gam;
    __builtin_nontemporal_store(x, (vf4*)out + (idx >> 2));
  }
}

extern "C" void kernel_launch(void* const* d_in, const int* in_sizes, int n_in,
                              void* d_out, int out_size, void* d_ws, size_t ws_size,
                              hipStream_t stream) {
  const float* feat  = (const float*)d_in[0];
  const float* lA    = (const float*)d_in[1];
  const float* lB    = (const float*)d_in[2];
  const int*   lab   = (const int*)d_in[3];
  const float* ranks = (const float*)d_in[4];

  // ws layout: [0..3] min-uint, [4..7] max-uint, [8..11] slope, [12..15] bias, [16..] dis
  unsigned* gmin  = (unsigned*)d_ws;
  unsigned* gmax  = gmin + NCLS;
  float* slope    = (float*)d_ws + 2 * NCLS;
  float* bias     = (float*)d_ws + 3 * NCLS;
  float* disbuf   = (float*)d_ws + 4 * NCLS;
  const bool has_dis = ws_size >= (size_t)(4 * NCLS) * 4 + (size_t)NVOX * 4;

  float* out = (float*)d_out;

  k_init<<<1, 32, 0, stream>>>(gmin, gmax);
  if (has_dis)
    k_dis_stats<true><<<K1_BLOCKS, 256, 0, stream>>>(lA, lB, lab, gmin, gmax, disbuf);
  else
    k_dis_stats<false><<<K1_BLOCKS, 256, 0, stream>>>(lA, lB, lab, gmin, gmax, disbuf);
  k_final<<<1, 32, 0, stream>>>(gmin, gmax, ranks, slope, bias);
  if (has_dis)
    k_apply<true><<<NVOX / 1024, 256, 0, stream>>>(feat, lA, lB, lab, disbuf, slope, bias, out);
  else
    k_apply<false><<<NVOX / 1024, 256, 0, stream>>>(feat, lA, lB, lab, disbuf, slope, bias, out);
}